// SpectralThresholdHead_16020228014178
// MI455X (gfx1250) — compile-verified
//
#include <hip/hip_runtime.h>
#include <cstdint>

// Problem constants (match reference)
#define D_MODEL 1024
#define BANDS   4
#define KTAPS   8
#define PAD     4
#define TDIM    4096
#define BATCH   4
#define EPSV    1e-6f

// Tiling
#define WC      128              // channel columns per tile (512 B rows -> 1 wave b128 per row)
#define TILE_T  128              // output time rows per tile (amortizes weight loads 2x)
#define HROWS   (TILE_T + 7)     // input rows needed: t0-4 .. t0+TILE_T+2
#define LROWS   (TILE_T + 8)     // padded LDS row count
#define RB      4                // output rows per register block (11-tap reuse)

typedef int   v4i __attribute__((ext_vector_type(4)));
typedef float v2f __attribute__((ext_vector_type(2)));

__global__ __launch_bounds__(256) void spectral_head_kernel(
    const float* __restrict__ x,
    const float* __restrict__ bank_w,
    const float* __restrict__ tau_raw,
    const float* __restrict__ mix_w,
    const float* __restrict__ mix_b,
    float* __restrict__ out)
{
    __shared__ float s[LROWS * WC];   // 136*128*4 = 68 KB

    const int c0 = blockIdx.x * WC;
    const int t0 = blockIdx.y * TILE_T;
    const int b  = blockIdx.z;

    const int tid  = threadIdx.x;
    const int lane = tid & 31;
    const int wid  = tid >> 5;

    // ---- Stage input tile rows [t0-4, t0+TILE_T+2] into LDS via async DMA ----
    // Each lane moves 16 B; one wave covers a full 128-float row per issue.
    for (int r = wid; r < HROWS; r += 8) {
        const int tg = t0 - PAD + r;
        float* lp = &s[r * WC + lane * 4];
        if (tg >= 0 && tg < TDIM) {
            const float* gp = x + ((size_t)b * TDIM + tg) * D_MODEL + c0 + lane * 4;
#if defined(__AMDGCN__)
  #if __has_builtin(__builtin_amdgcn_global_load_async_to_lds_b128)
            __builtin_amdgcn_global_load_async_to_lds_b128(
                (v4i*)(uintptr_t)gp, (v4i*)(uintptr_t)lp, 0, 0);
  #else
            unsigned lds_byte = (unsigned)(uintptr_t)lp;
            asm volatile("global_load_async_to_lds_b128 %0, %1, off"
                         :: "v"(lds_byte), "v"(gp)
                         : "memory");
  #endif
#else
            *(float4*)lp = *(const float4*)gp;
#endif
        } else {
            *(float4*)lp = make_float4(0.f, 0.f, 0.f, 0.f);
        }
    }

#if defined(__AMDGCN__)
  #if __has_builtin(__builtin_amdgcn_s_wait_asynccnt)
    __builtin_amdgcn_s_wait_asynccnt(0);
  #else
    asm volatile("s_wait_asynccnt 0" ::: "memory");
  #endif
#endif
    __syncthreads();

    // ---- Per-thread constants (channel c fixed per thread) ----
    const int col = tid & (WC - 1);     // 0..127
    const int rg  = tid >> 7;           // 0..1 (row group)
    const int c   = c0 + col;

    // basis packed over band pairs: b01[k] = (basis[0][k], basis[1][k]), etc.
    v2f b01[KTAPS], b23[KTAPS];
    {
        const float* bw = bank_w + (size_t)c * BANDS * KTAPS;  // row o = c*BANDS+n
        #pragma unroll
        for (int k = 0; k < KTAPS; ++k) {
            b01[k] = (v2f){bw[0 * KTAPS + k], bw[1 * KTAPS + k]};
            b23[k] = (v2f){bw[2 * KTAPS + k], bw[3 * KTAPS + k]};
        }
    }

    float tau[BANDS], mw[BANDS];
    #pragma unroll
    for (int n = 0; n < BANDS; ++n) {
        tau[n] = log1pf(expf(tau_raw[n]));                 // softplus
        mw[n]  = mix_w[(size_t)c * BANDS + n];
    }
    const float mb = mix_b[c];

    // ---- Compute: each thread does TILE_T/2 consecutive t rows for one channel,
    //      in register blocks of 4 rows sharing an 11-tap window. ----
    for (int j4 = 0; j4 < TILE_T / 2; j4 += RB) {
        const int base = rg * (TILE_T / 2) + j4;           // LDS row of first tap

        float w[RB + KTAPS - 1];                           // 11 taps
        #pragma unroll
        for (int k = 0; k < RB + KTAPS - 1; ++k)
            w[k] = s[(base + k) * WC + col];

        #pragma unroll
        for (int i = 0; i < RB; ++i) {
            v2f y01 = (v2f){0.f, 0.f};
            v2f y23 = (v2f){0.f, 0.f};
            #pragma unroll
            for (int k = 0; k < KTAPS; ++k) {
                const v2f ws = (v2f){w[i + k], w[i + k]};  // tap broadcast
                y01 = __builtin_elementwise_fma(b01[k], ws, y01);  // v_pk_fma_f32
                y23 = __builtin_elementwise_fma(b23[k], ws, y23);
            }
            const float y[BANDS] = {y01.x, y01.y, y23.x, y23.y};

            float acc = mb;
            #pragma unroll
            for (int n = 0; n < BANDS; ++n) {
                const float a = fabsf(y[n]);
#if defined(__AMDGCN__) && __has_builtin(__builtin_amdgcn_rcpf)
                const float r = __builtin_amdgcn_rcpf(a + EPSV);   // v_rcp_f32
#else
                const float r = 1.0f / (a + EPSV);
#endif
                const float th = fmaxf(a - tau[n], 0.f);
                acc = fmaf(th * (y[n] * r), mw[n], acc);
            }

            const int t = t0 + base + i;                   // global output row
            float* op = out + ((size_t)b * TDIM + t) * D_MODEL + c;
#if defined(__AMDGCN__)
            __builtin_nontemporal_store(acc, op);          // streaming store
#else
            *op = acc;
#endif
        }
    }
}

extern "C" void kernel_launch(void* const* d_in, const int* in_sizes, int n_in,
                              void* d_out, int out_size, void* d_ws, size_t ws_size,
                              hipStream_t stream) {
    const float* x   = (const float*)d_in[0];
    const float* bw  = (const float*)d_in[1];
    const float* tr  = (const float*)d_in[2];
    const float* mw  = (const float*)d_in[3];
    const float* mb  = (const float*)d_in[4];
    float* out = (float*)d_out;

    dim3 grid(D_MODEL / WC, TDIM / TILE_T, BATCH);   // (8, 32, 4)
    spectral_head_kernel<<<grid, 256, 0, stream>>>(x, bw, tr, mw, mb, out);
}